// GATCon_3985729650708
// MI455X (gfx1250) — compile-verified
//
#include <hip/hip_runtime.h>
#include <hip/hip_bf16.h>
#include <math.h>

// ---------------- problem constants (match reference) ----------------
static constexpr int Bg  = 512;          // graphs
static constexpr int NGc = 128;          // nodes / graph
static constexpr int EGc = 1024;         // edges / graph
static constexpr int Nn  = Bg * NGc;     // 65536 nodes
static constexpr int Ee  = Bg * EGc;     // 524288 edges
static constexpr int F   = 128;          // H*C
static constexpr int KEc = 204;          // int(EG*0.2)
static constexpr int KNc = 25;           // int(NG*0.2)
static constexpr int Kin = 78;           // input feature dim
#define LRELU_SLOPE 0.2f

typedef __attribute__((ext_vector_type(2))) float v2f;
typedef __attribute__((ext_vector_type(8))) float v8f;

// monotone float<->uint encoding so unsigned atomicMax == float max
__device__ __forceinline__ unsigned fenc(float f) {
  unsigned u = __float_as_uint(f);
  return (u & 0x80000000u) ? ~u : (u | 0x80000000u);
}
__device__ __forceinline__ float fdec(unsigned u) {
  unsigned v = (u & 0x80000000u) ? (u & 0x7FFFFFFFu) : ~u;
  return __uint_as_float(v);
}
__device__ __forceinline__ float lrelu(float z) { return z > 0.0f ? z : LRELU_SLOPE * z; }

// ---------------- generic fp32 WMMA GEMM: C = act(A@B + bias) ----------------
// A: MxK row-major, B: KxN row-major, C: MxN row-major. M,N multiples of 16.
// One wave per 16x16 output tile, V_WMMA_F32_16X16X4_F32 accumulation.
// Main K-loop is provably in-bounds (no guards, A pair loaded as one b64);
// K%4 tail uses clamped addresses + lane selects (v_cndmask, EXEC untouched).
__global__ void gemm_wmma_f32(const float* __restrict__ A, const float* __restrict__ Bm,
                              const float* __restrict__ bias, float* __restrict__ C,
                              int M, int N, int K, int act) {
  const int lane  = threadIdx.x & 31;
  const int wave  = threadIdx.x >> 5;
  const int tilesN = N >> 4;
  const int tilesM = M >> 4;
  const int tile = blockIdx.x * (blockDim.x >> 5) + wave;
  if (tile >= tilesM * tilesN) return;            // wave-uniform: EXEC stays all-ones
  const int tm = tile / tilesN;
  const int tn = tile % tilesN;
  const int r15  = lane & 15;
  const int koff = (lane >> 4) << 1;              // lanes 16..31 hold K+2,K+3
  const int col  = tn * 16 + r15;
  const float* __restrict__ Ap = A + (long)(tm * 16 + r15) * K + koff;  // 8B aligned
  const float* __restrict__ Bp = Bm + (long)koff * N + col;

  const int Kmain = K & ~3;
  v8f acc = {};
  int k0 = 0;
#pragma unroll 4
  for (; k0 < Kmain; k0 += 4) {                   // ka+1 <= Kmain-1 < K: no guards
    const float2 av = *(const float2*)(Ap + k0);  // global_load_b64
    v2f a, b;
    a.x = av.x;
    a.y = av.y;
    b.x = Bp[(long)k0 * N];
    b.y = Bp[(long)(k0 + 1) * N];
    acc = __builtin_amdgcn_wmma_f32_16x16x4_f32(false, a, false, b, (short)0, acc,
                                                false, false);
  }
  if (k0 < K) {                                    // K%4 tail, branch-free
    const int ka  = k0 + koff;
    const int ka0 = (ka     < K) ? ka     : K - 1; // clamp: always a valid address
    const int ka1 = (ka + 1 < K) ? ka + 1 : K - 1;
    const long ar = (long)(tm * 16 + r15) * K;
    const float a0 = A[ar + ka0];
    const float a1 = A[ar + ka1];
    const float b0 = Bm[(long)ka0 * N + col];
    const float b1 = Bm[(long)ka1 * N + col];
    v2f a, b;
    a.x = (ka     < K) ? a0 : 0.0f;                // v_cndmask, no EXEC games
    a.y = (ka + 1 < K) ? a1 : 0.0f;
    b.x = (ka     < K) ? b0 : 0.0f;
    b.y = (ka + 1 < K) ? b1 : 0.0f;
    acc = __builtin_amdgcn_wmma_f32_16x16x4_f32(false, a, false, b, (short)0, acc,
                                                false, false);
  }
  const float bv = bias ? bias[col] : 0.0f;
  const int mbase = tm * 16 + ((lane >> 4) << 3);  // +8 for upper half-lanes
#pragma unroll
  for (int r = 0; r < 8; ++r) {
    float v = acc[r] + bv;
    if (act == 1) v = fmaxf(v, 0.0f);
    C[(long)(mbase + r) * N + col] = v;
  }
}

// ---------------- es/ed: per-(node,head) dot of h with a_src/a_dst ----------------
__global__ void es_ed_kernel(const float* __restrict__ h, const float* __restrict__ a_src,
                             const float* __restrict__ a_dst,
                             float* __restrict__ es, float* __restrict__ ed) {
  const int idx = blockIdx.x * blockDim.x + threadIdx.x;   // node*4 + head
  if (idx >= Nn * 4) return;
  const int n = idx >> 2, hh = idx & 3;
  const float* hp = h + (long)n * F + hh * 32;
  const float* as = a_src + hh * 32;
  const float* ad = a_dst + hh * 32;
  float s = 0.0f, d = 0.0f;
#pragma unroll 8
  for (int c = 0; c < 32; ++c) { const float v = hp[c]; s += v * as[c]; d += v * ad[c]; }
  es[idx] = s; ed[idx] = d;
}

__global__ void fill_zero_u32(unsigned* __restrict__ p, long n) {
  const long i = (long)blockIdx.x * blockDim.x + threadIdx.x;
  if (i < n) p[i] = 0u;
}

// ---------------- leaky-relu logits + segment max over dst ----------------
__global__ void logits_max(const int* __restrict__ src, const int* __restrict__ dst,
                           const float* __restrict__ es, const float* __restrict__ ed,
                           const int* __restrict__ keep, const float* __restrict__ mask,
                           float* __restrict__ logit, unsigned* __restrict__ mmax) {
  const long idx = (long)blockIdx.x * blockDim.x + threadIdx.x;
  if (idx >= (long)(Ee + Nn) * 4) return;
  const int e = (int)(idx >> 2), hh = (int)(idx & 3);
  int d; float l;
  if (e < Ee) {
    const int s = src[e]; d = dst[e];
    if (keep && !keep[e]) {
      l = -__builtin_inff();                      // deleted edge == no contribution
    } else {
      float vs = es[(long)s * 4 + hh];
      float vd = ed[(long)d * 4 + hh];
      if (mask) { vs *= mask[s]; vd *= mask[d]; }
      l = lrelu(vs + vd);
    }
  } else {                                         // self loop
    d = e - Ee;
    float z = es[(long)d * 4 + hh] + ed[(long)d * 4 + hh];
    if (mask) z *= mask[d];
    l = lrelu(z);
  }
  logit[idx] = l;
  atomicMax(&mmax[(long)d * 4 + hh], fenc(l));
}

// ---------------- exp(logit - max) + segment sum ----------------
__global__ void exp_den(const int* __restrict__ dst, const unsigned* __restrict__ mmax,
                        float* __restrict__ logit_ex, float* __restrict__ den) {
  const long idx = (long)blockIdx.x * blockDim.x + threadIdx.x;
  if (idx >= (long)(Ee + Nn) * 4) return;
  const int e = (int)(idx >> 2), hh = (int)(idx & 3);
  const int d = (e < Ee) ? dst[e] : (e - Ee);
  const float m = fdec(mmax[(long)d * 4 + hh]);
  const float ex = expf(logit_ex[idx] - m);        // -inf -> 0 for deleted edges
  logit_ex[idx] = ex;
  atomicAdd(&den[(long)d * 4 + hh], ex);
}

// ---------------- alpha + scatter aggregation: one wave per edge ----------------
__global__ void alpha_agg(const int* __restrict__ src, const int* __restrict__ dst,
                          const float* __restrict__ exb, const float* __restrict__ den,
                          const float* __restrict__ h, const float* __restrict__ mask,
                          float* __restrict__ outAgg,
                          float* __restrict__ ewOut, float* __restrict__ xwOut) {
  const int wid  = (int)(((long)blockIdx.x * blockDim.x + threadIdx.x) >> 5);
  const int lane = threadIdx.x & 31;
  if (wid >= Ee + Nn) return;
  int s, d;
  if (wid < Ee) { s = src[wid]; d = dst[wid]; } else { s = d = wid - Ee; }
  if (ewOut && lane < 4) {                         // pass 1: emit alpha (ew1/xw1)
    const float a = exb[(long)wid * 4 + lane] / (den[(long)d * 4 + lane] + 1e-16f);
    if (wid < Ee) ewOut[(long)wid * 4 + lane] = a;
    else          xwOut[(long)(wid - Ee) * 4 + lane] = a;
  }
  const int hh = lane >> 3;                        // 4 channels per lane, one head
  const float a  = exb[(long)wid * 4 + hh] / (den[(long)d * 4 + hh] + 1e-16f);
  const float ms = mask ? mask[s] : 1.0f;
  const float4 hv = *(const float4*)(h + (long)s * F + lane * 4);
  float* op = outAgg + (long)d * F + lane * 4;
  atomicAdd(op + 0, a * ms * hv.x);
  atomicAdd(op + 1, a * ms * hv.y);
  atomicAdd(op + 2, a * ms * hv.z);
  atomicAdd(op + 3, a * ms * hv.w);
}

// ---------------- contiguous-segment mean pool (batch = repeat(arange(B),NG)) ----------------
__global__ void pool_mean(const float* __restrict__ nodeFeat, float* __restrict__ pooled) {
  const int b = blockIdx.x, c = threadIdx.x;       // 128 threads = features
  const float* p = nodeFeat + (long)b * NGc * F + c;
  float sum = 0.0f;
  for (int n = 0; n < NGc; ++n) sum += p[(long)n * F];
  pooled[(long)b * F + c] = sum * (1.0f / NGc);
}

// ---------------- per-graph top-KE edge deletion (exact stable descending rank) ----------------
__global__ void edge_del(const float* __restrict__ ewAlpha /*E x 4*/, int* __restrict__ keep) {
  __shared__ float w[EGc];
  const int b = blockIdx.x;
  const float* base = ewAlpha + (long)b * EGc * 4;
  for (int i = threadIdx.x; i < EGc; i += blockDim.x)
    w[i] = (base[i*4] + base[i*4+1] + base[i*4+2] + base[i*4+3]) * 0.25f;
  __syncthreads();
  for (int i = threadIdx.x; i < EGc; i += blockDim.x) {
    const float wi = w[i];
    int rank = 0;
    for (int j = 0; j < EGc; ++j) {
      const float wj = w[j];
      rank += (wj > wi) || (wj == wi && j < i);    // stable argsort(-w) position
    }
    keep[(long)b * EGc + i] = (rank >= KEc) ? 1 : 0;
  }
}

// ---------------- per-graph top-KN node zeroing -> float mask ----------------
__global__ void node_del(const float* __restrict__ xwAlpha /*N x 4*/, float* __restrict__ mask) {
  __shared__ float w[NGc];
  const int b = blockIdx.x;
  const float* base = xwAlpha + (long)b * NGc * 4;
  for (int i = threadIdx.x; i < NGc; i += blockDim.x)
    w[i] = (base[i*4] + base[i*4+1] + base[i*4+2] + base[i*4+3]) * 0.25f;
  __syncthreads();
  for (int i = threadIdx.x; i < NGc; i += blockDim.x) {
    const float wi = w[i];
    int rank = 0;
    for (int j = 0; j < NGc; ++j) {
      const float wj = w[j];
      rank += (wj > wi) || (wj == wi && j < i);
    }
    mask[(long)b * NGc + i] = (rank < KNc) ? 0.0f : 1.0f;
  }
}

// ---------------- host-side orchestration ----------------
static inline int cdiv(long a, long b) { return (int)((a + b - 1) / b); }

extern "C" void kernel_launch(void* const* d_in, const int* in_sizes, int n_in,
                              void* d_out, int out_size, void* d_ws, size_t ws_size,
                              hipStream_t stream) {
  (void)in_sizes; (void)n_in; (void)out_size; (void)ws_size;
  const float* x      = (const float*)d_in[0];
  const int*   eidx   = (const int*)  d_in[1];
  const float* W      = (const float*)d_in[3];
  const float* a_src  = (const float*)d_in[4];
  const float* a_dst  = (const float*)d_in[5];
  const float* W1     = (const float*)d_in[6];
  const float* b1     = (const float*)d_in[7];
  const float* W2     = (const float*)d_in[8];
  const float* b2     = (const float*)d_in[9];
  const float* W3     = (const float*)d_in[10];
  const float* b3     = (const float*)d_in[11];
  const int* src = eidx;
  const int* dst = eidx + Ee;

  // ---- output layout: x1 | out1 | x2 | out2 | ew1 | xw1 ----
  float* out  = (float*)d_out;
  float* x1   = out;
  float* out1 = out + (long)Bg * F;
  float* x2   = out + 2L * Bg * F;
  float* out2 = out + 3L * Bg * F;
  float* ew1  = out + 4L * Bg * F;
  float* xw1  = ew1 + (long)Ee * 4;

  // ---- workspace bump allocator (all sizes multiples of 256B) ----
  char* wp = (char*)d_ws;
  auto alloc = [&](size_t bytes) { void* p = wp; wp += (bytes + 255) & ~(size_t)255; return p; };
  float*    h    = (float*)   alloc((size_t)Nn * F * 4);
  float*    es   = (float*)   alloc((size_t)Nn * 4 * 4);
  float*    ed   = (float*)   alloc((size_t)Nn * 4 * 4);
  unsigned* mmax = (unsigned*)alloc((size_t)Nn * 4 * 4);   // zero-zone begin
  float*    den  = (float*)   alloc((size_t)Nn * 4 * 4);
  float*    agg  = (float*)   alloc((size_t)Nn * F * 4);   // zero-zone end (contiguous)
  float*    exb  = (float*)   alloc((size_t)(Ee + Nn) * 4 * 4);
  float*    z1   = (float*)   alloc((size_t)Bg * 1024 * 4);
  float*    z2   = (float*)   alloc((size_t)Bg * 512 * 4);
  int*      keep = (int*)     alloc((size_t)Ee * 4);
  float*    mask = (float*)   alloc((size_t)Nn * 4);
  const long zeroZoneU32 = (long)Nn * 4 + (long)Nn * 4 + (long)Nn * F;

  const long tot4 = (long)(Ee + Nn) * 4;

  // ===== h = x @ W  (65536x78 @ 78x128, fp32 WMMA) — shared by both passes =====
  {
    const int tiles = (Nn / 16) * (F / 16);
    gemm_wmma_f32<<<cdiv(tiles, 8), 256, 0, stream>>>(x, W, nullptr, h, Nn, F, Kin, 0);
  }
  es_ed_kernel<<<cdiv((long)Nn * 4, 256), 256, 0, stream>>>(h, a_src, a_dst, es, ed);

  // ===== pass 1: attention softmax + aggregation =====
  fill_zero_u32<<<cdiv(zeroZoneU32, 256), 256, 0, stream>>>(mmax, zeroZoneU32);
  logits_max<<<cdiv(tot4, 256), 256, 0, stream>>>(src, dst, es, ed, nullptr, nullptr, exb, mmax);
  exp_den<<<cdiv(tot4, 256), 256, 0, stream>>>(dst, mmax, exb, den);
  alpha_agg<<<cdiv((long)(Ee + Nn) * 32, 256), 256, 0, stream>>>(
      src, dst, exb, den, h, nullptr, agg, ew1, xw1);
  pool_mean<<<Bg, F, 0, stream>>>(agg, x1);

  // ===== out1 = MLP(x1) =====
  gemm_wmma_f32<<<cdiv((Bg/16)*(1024/16), 8), 256, 0, stream>>>(x1, W1, b1, z1, Bg, 1024, 128, 1);
  gemm_wmma_f32<<<cdiv((Bg/16)*(512/16),  8), 256, 0, stream>>>(z1, W2, b2, z2, Bg, 512, 1024, 1);
  gemm_wmma_f32<<<cdiv((Bg/16)*(128/16),  8), 256, 0, stream>>>(z2, W3, b3, out1, Bg, 128, 512, 0);

  // ===== attention_del: top-20% edges removed, top-20% nodes zeroed =====
  edge_del<<<Bg, 256, 0, stream>>>(ew1, keep);
  node_del<<<Bg, 128, 0, stream>>>(xw1, mask);

  // ===== pass 2: reuse h/es/ed with mask+keep (x_del@W == mask ⊙ h) =====
  fill_zero_u32<<<cdiv(zeroZoneU32, 256), 256, 0, stream>>>(mmax, zeroZoneU32);
  logits_max<<<cdiv(tot4, 256), 256, 0, stream>>>(src, dst, es, ed, keep, mask, exb, mmax);
  exp_den<<<cdiv(tot4, 256), 256, 0, stream>>>(dst, mmax, exb, den);
  alpha_agg<<<cdiv((long)(Ee + Nn) * 32, 256), 256, 0, stream>>>(
      src, dst, exb, den, h, mask, agg, nullptr, nullptr);
  pool_mean<<<Bg, F, 0, stream>>>(agg, x2);

  // ===== out2 = MLP(x2) =====
  gemm_wmma_f32<<<cdiv((Bg/16)*(1024/16), 8), 256, 0, stream>>>(x2, W1, b1, z1, Bg, 1024, 128, 1);
  gemm_wmma_f32<<<cdiv((Bg/16)*(512/16),  8), 256, 0, stream>>>(z1, W2, b2, z2, Bg, 512, 1024, 1);
  gemm_wmma_f32<<<cdiv((Bg/16)*(128/16),  8), 256, 0, stream>>>(z2, W3, b3, out2, Bg, 128, 512, 0);
}